// OuterProductNetwork2_75273596829806
// MI455X (gfx1250) — compile-verified
//
#include <hip/hip_runtime.h>

// ---------------------------------------------------------------------------
// Reference: x[16384, 32, 64] f32 -> s[b,f] = sum_e x[b,f,e];
//            out[b,p] = s[b, ROW[p]] * s[b, COL[p]], P = 496 pairs.
// Bandwidth-bound (~160 MiB @ 23.3 TB/s). Row-sum done as S = X * ones via
// V_WMMA_F32_16X16X4_F32 (full f32 precision), fed from bank-conflict-free LDS.
// ---------------------------------------------------------------------------

typedef __attribute__((ext_vector_type(2))) float v2f;
typedef __attribute__((ext_vector_type(8))) float v8f;

#define NBATCH      16384
#define NFIELD      32
#define NEMBED      64
#define NPAIR       496
#define BB          4                  // batches per block
#define ROWS        (BB * NFIELD)      // 128 rows of 64 floats per block
#define LDS_STRIDE  68                 // dwords/row: 4*m+{0..3} banking -> conflict-free
#define THREADS     256
#define OUT_PER_BLK (BB * NPAIR)       // 1984

struct PairTab { unsigned char r[NPAIR]; unsigned char c[NPAIR]; };
static constexpr PairTab make_tab() {
    PairTab t{};
    int p = 0;
    for (int i = 0; i < NFIELD - 1; ++i)
        for (int j = i + 1; j < NFIELD; ++j) {
            t.r[p] = (unsigned char)i;
            t.c[p] = (unsigned char)j;
            ++p;
        }
    return t;
}
__constant__ PairTab TAB = make_tab();

__global__ __launch_bounds__(THREADS)
void opn_kernel(const float* __restrict__ x, float* __restrict__ out) {
    __shared__ float tile[ROWS * LDS_STRIDE];   // 128 * 68 * 4 = 34,816 B
    __shared__ float s[ROWS];                   // 128 row sums

    const int tid = threadIdx.x;
    const int blk = blockIdx.x;

    // ---- Stage 1: coalesced global -> LDS (padded rows) -------------------
    // Block reads a contiguous 32 KB slab; each wave instruction covers 512 B.
    const float4* __restrict__ src =
        (const float4*)(x + (size_t)blk * (size_t)(ROWS * NEMBED));
#pragma unroll
    for (int k = 0; k < (ROWS * NEMBED / 4) / THREADS; ++k) {   // 8 iters
        const int fi   = tid + k * THREADS;   // float4 index within slab
        const int row  = fi >> 4;             // 16 float4s per 64-float row
        const int col  = (fi & 15) << 2;      // float offset within row
        const float4 v = src[fi];
        *(float4*)&tile[row * LDS_STRIDE + col] = v;   // ds_store_b128
    }
    __syncthreads();

    // ---- Stage 2: WMMA row-sum, one 16-row tile per wave ------------------
    // A-layout (32-bit A 16x4): lanes 0-15 hold K={k0,k0+1}, lanes 16-31 hold
    // K={k0+2,k0+3}, M = lane%16.  B = ones => D[m,n] = rowsum(m) for all n.
    const int wave  = tid >> 5;
    const int lane  = tid & 31;
    const int m     = lane & 15;
    const int khalf = (lane >> 4) << 1;       // 0 or 2
    const float* __restrict__ arow = &tile[(wave * 16 + m) * LDS_STRIDE + khalf];

    v2f ones; ones[0] = 1.0f; ones[1] = 1.0f;
    v8f acc = {};
#pragma unroll
    for (int k0 = 0; k0 < NEMBED; k0 += 4) {
        v2f a;
        a[0] = arow[k0];          // one ds_load_b64, conflict-free (stride 68)
        a[1] = arow[k0 + 1];
        acc = __builtin_amdgcn_wmma_f32_16x16x4_f32(
                  /*neg_a=*/false, a, /*neg_b=*/false, ones,
                  /*c_mod=*/(short)0, acc, /*reuse_a=*/false, /*reuse_b=*/false);
    }

    // D layout: VGPR v, lanes 0-15 -> (M=v, N=lane); lanes 16-31 -> (M=v+8).
    // Column N=0 lives in lane 0 (M=0..7) and lane 16 (M=8..15).
    if (m == 0) {
        const int rbase = wave * 16 + ((lane >> 4) ? 8 : 0);
#pragma unroll
        for (int v = 0; v < 8; ++v) s[rbase + v] = acc[v];
    }
    __syncthreads();

    // ---- Stage 3: pairwise products, contiguous coalesced stores ----------
    float* __restrict__ obase = out + (size_t)blk * OUT_PER_BLK;
    for (int idx = tid; idx < OUT_PER_BLK; idx += THREADS) {
        const int bl = idx / NPAIR;
        const int p  = idx - bl * NPAIR;
        const float a0 = s[bl * NFIELD + TAB.r[p]];
        const float b0 = s[bl * NFIELD + TAB.c[p]];
        obase[idx] = a0 * b0;
    }
}

extern "C" void kernel_launch(void* const* d_in, const int* in_sizes, int n_in,
                              void* d_out, int out_size, void* d_ws, size_t ws_size,
                              hipStream_t stream) {
    (void)in_sizes; (void)n_in; (void)d_ws; (void)ws_size; (void)out_size;
    const float* x = (const float*)d_in[0];
    float* out = (float*)d_out;
    dim3 grid(NBATCH / BB);      // 4096 blocks
    dim3 block(THREADS);         // 8 waves (wave32)
    opn_kernel<<<grid, block, 0, stream>>>(x, out);
}